// PostProcess_2113123909850
// MI455X (gfx1250) — compile-verified
//
#include <hip/hip_runtime.h>
#include <hip/hip_bf16.h>
#include <stdint.h>

// Problem constants (match reference)
#define BB   256
#define QQ   900
#define CC   91
#define KK   300
#define QC   (QQ * CC)      // 81900
#define QC4  (QC / 4)       // 20475 float4 per batch row (exactly divisible)
#define NBIN 1024
#define CAP  4096
#define KPAD 304            // 19 * 16
#define NTILE 19
#define SIGMA_INV 2.0f      // 1/0.5
#define THR  0.001f
#define BK   (BB * KK)

typedef __attribute__((ext_vector_type(2))) float v2f;
typedef __attribute__((ext_vector_type(8))) float v8f;

// ---- float <-> totally-ordered uint key (ascending key == ascending float) ----
__device__ __forceinline__ uint32_t fkey(float f) {
  uint32_t u = __float_as_uint(f);
  return (u & 0x80000000u) ? ~u : (u | 0x80000000u);
}
__device__ __forceinline__ float funkey(uint32_t k) {
  uint32_t u = (k & 0x80000000u) ? (k & 0x7fffffffu) : ~k;
  return __uint_as_float(u);
}

// CDNA5 async global->LDS copy (ASYNCcnt-tracked), per-lane 16B
__device__ __forceinline__ void async_ld_b128(uint32_t ldsoff, const void* g) {
  asm volatile("global_load_async_to_lds_b128 %0, %1, off"
               :: "v"(ldsoff), "v"(g) : "memory");
}
__device__ __forceinline__ void wait_asynccnt0() {
  asm volatile("s_wait_asynccnt 0" ::: "memory");
}
__device__ __forceinline__ void wait_asynccnt1() {
  asm volatile("s_wait_asynccnt 1" ::: "memory");
}
__device__ __forceinline__ void wait_dscnt0() {
  asm volatile("s_wait_dscnt 0" ::: "memory");
}

// =====================================================================
// Kernel A: per-batch sigmoid-monotone top-K via histogram threshold.
// One block per batch. Pass 1 streams from HBM through double-buffered
// async->LDS tiles; pass 2 re-reads from L2 (84 MB total < 192 MB L2).
// =====================================================================
__global__ __launch_bounds__(512) void topk_kernel(
    const float* __restrict__ logits,
    float* __restrict__ out_labels,   // d_out + BK
    float* __restrict__ ws_scores,    // [BK]
    int*   __restrict__ ws_qidx)      // [BK]
{
  __shared__ uint32_t hist[NBIN];
  __shared__ __align__(16) float stage[2][512 * 4];
  __shared__ uint32_t ckey[CAP];
  __shared__ uint32_t cidx[CAP];
  __shared__ uint32_t s_thr, s_cnt;

  const int b   = blockIdx.x;
  const int tid = threadIdx.x;
  const float* lb = logits + (size_t)b * QC;

  for (int i = tid; i < NBIN; i += 512) hist[i] = 0u;
  for (int i = tid; i < CAP; i += 512) { ckey[i] = 0u; cidx[i] = 0u; }
  if (tid == 0) s_cnt = 0u;
  __syncthreads();

  // ---- pass 1: histogram; async staging, double buffered (1 tile in flight) ----
  const uint32_t loff0 = (uint32_t)(uintptr_t)(&stage[0][tid * 4]);
  const uint32_t loff1 = (uint32_t)(uintptr_t)(&stage[1][tid * 4]);
  const int ntiles = (QC4 + 511) / 512;   // 40; last tile: threads 0..506 valid,
                                          // so EVERY wave issues in every tile.
  if (tid < QC4) async_ld_b128(loff0, lb + (size_t)tid * 4);
  for (int t = 0; t < ntiles; ++t) {
    const uint32_t lcur = (t & 1) ? loff1 : loff0;
    const uint32_t lnxt = (t & 1) ? loff0 : loff1;
    const int gn = (t + 1) * 512 + tid;
    if (t + 1 < ntiles && gn < QC4) async_ld_b128(lnxt, lb + (size_t)gn * 4);
    if (t + 1 < ntiles) wait_asynccnt1();   // oldest (current tile) complete
    else               wait_asynccnt0();
    const int g = t * 512 + tid;
    if (g < QC4) {
      float4 v = *(const float4*)(const void*)(uintptr_t)lcur;
      float vv[4] = { v.x, v.y, v.z, v.w };
#pragma unroll
      for (int c = 0; c < 4; ++c)
        atomicAdd(&hist[fkey(vv[c]) >> 22], 1u);
    }
  }
  __syncthreads();

  // ---- suffix scan: largest bin t with count(key-bin >= t) >= K ----
  if (tid == 0) {
    uint32_t run = 0, t = 0;
    for (int v = NBIN - 1; v >= 0; --v) {
      run += hist[v];
      if (run >= KK) { t = (uint32_t)v; break; }
    }
    s_thr = t;
  }
  __syncthreads();
  const uint32_t thr = s_thr;

  // ---- pass 2: compact candidates (streaming b128 from L2) ----
  for (int g = tid; g < QC4; g += 512) {
    float4 v = ((const float4*)lb)[g];
    float vv[4] = { v.x, v.y, v.z, v.w };
#pragma unroll
    for (int c = 0; c < 4; ++c) {
      uint32_t k = fkey(vv[c]);
      if ((k >> 22) >= thr) {
        uint32_t pos = atomicAdd(&s_cnt, 1u);
        if (pos < CAP) { ckey[pos] = k; cidx[pos] = (uint32_t)(g * 4 + c); }
      }
    }
  }
  __syncthreads();

  // ---- bitonic sort (descending by key, idx payload) over fixed CAP ----
  for (uint32_t kk = 2; kk <= CAP; kk <<= 1) {
    for (uint32_t j = kk >> 1; j > 0; j >>= 1) {
      for (uint32_t i = tid; i < CAP; i += 512) {
        uint32_t l = i ^ j;
        if (l > i) {
          uint32_t a = ckey[i], c = ckey[l];
          bool desc = ((i & kk) == 0);
          if (desc ? (a < c) : (a > c)) {
            ckey[i] = c; ckey[l] = a;
            uint32_t ta = cidx[i]; cidx[i] = cidx[l]; cidx[l] = ta;
          }
        }
      }
      __syncthreads();
    }
  }

  // ---- emit top-K: sigmoid(score), label, q-index ----
  if (tid < KK) {
    float logit = funkey(ckey[tid]);
    float sc = 1.0f / (1.0f + expf(-logit));
    uint32_t flat = cidx[tid];
    int q = (int)(flat / CC);
    int label = (int)(flat % CC);
    size_t o = (size_t)b * KK + tid;
    out_labels[o] = (float)label;
    ws_scores[o]  = sc;
    ws_qidx[o]    = q;
  }
}

// =====================================================================
// Kernel B: box transform via V_WMMA_F32_16X16X4_F32 + sequential
// soft-NMS executed by a single wave (wave-local s_wait_dscnt fences,
// no block barriers in the 300-iteration loop). One block per batch.
// =====================================================================
__global__ __launch_bounds__(512) void nms_kernel(
    const float* __restrict__ pred_boxes,   // [B,Q,4] cxcywh
    const float* __restrict__ tsizes,       // [B,2]  (h, w)
    const float* __restrict__ ws_scores,
    const int*   __restrict__ ws_qidx,
    float* __restrict__ out)                // full d_out base
{
  __shared__ float abox[KPAD][4];           // gathered cxcywh
  __shared__ float xyxy[4][KPAD];           // SoA x1,y1,x2,y2 (scaled)
  __shared__ float ss[KPAD];

  const int b    = blockIdx.x;
  const int tid  = threadIdx.x;
  const int wave = tid >> 5;
  const int lane = tid & 31;

  const float imh = tsizes[b * 2 + 0];
  const float imw = tsizes[b * 2 + 1];

  for (int i = tid; i < KPAD; i += 512) {
    if (i < KK) {
      int q = ws_qidx[(size_t)b * KK + i];
      const float* p = pred_boxes + ((size_t)b * QQ + q) * 4;
      abox[i][0] = p[0]; abox[i][1] = p[1]; abox[i][2] = p[2]; abox[i][3] = p[3];
      ss[i] = ws_scores[(size_t)b * KK + i];
    } else {
      abox[i][0] = abox[i][1] = abox[i][2] = abox[i][3] = 0.f;
      ss[i] = 0.f;
    }
  }
  __syncthreads();

  // xyxy_scaled = A(16x4 cxcywh) * W(4x16); W folds cxcywh->xyxy and the
  // per-image (w,h,w,h) scaling. Full-F32 WMMA => exact.
  // W rows: k=0: imw at j=0,2 | k=1: imh at j=1,3
  //         k=2: -+0.5*imw at j=0,2 | k=3: -+0.5*imh at j=1,3
  for (int t = wave; t < NTILE; t += 16) {       // wave-uniform: EXEC all-1s
    const int half  = lane >> 4;                  // 0: K=0,1  1: K=2,3
    const int m     = lane & 15;                  // A row
    const int n     = lane & 15;                  // B/D column
    const int row   = t * 16 + m;

    v2f a; a.x = abox[row][half * 2]; a.y = abox[row][half * 2 + 1];

    // Branch-free W entries (two v_cndmask chains, no control flow):
    const float mag = half ? ((n & 2) ? 0.5f : -0.5f) : 1.0f;
    const bool  j4  = (n < 4);
    v2f bm;
    bm.x = (j4 && !(n & 1)) ? mag * imw : 0.f;    // even-k row -> even j
    bm.y = (j4 &&  (n & 1)) ? mag * imh : 0.f;    // odd-k row  -> odd j
    v8f acc = {};
    acc = __builtin_amdgcn_wmma_f32_16x16x4_f32(
        /*neg_a=*/false, a, /*neg_b=*/false, bm,
        /*c_mod=*/(short)0, acc, /*reuse_a=*/false, /*reuse_b=*/false);

    // D layout: vgpr v, lane l -> D[v + (l>>4)*8][l&15]; arithmetic SoA store
    if (j4) {
      float* dst = &xyxy[n][t * 16 + half * 8];
#pragma unroll
      for (int v = 0; v < 8; ++v) dst[v] = acc[v];
    }
  }
  __syncthreads();

  float* bx1 = xyxy[0]; float* by1 = xyxy[1];
  float* bx2 = xyxy[2]; float* by2 = xyxy[3];

  // ---- soft-NMS: single wave runs the 300-step sequential loop ----
  if (tid < 32) {
    for (int i = 0; i < KK; ++i) {
      wait_dscnt0();
      // argmax over ss[i..K), first-index tie-break
      float bv = -1e30f; int bi = i;
      for (int j = i + lane; j < KK; j += 32) {
        float v = ss[j];
        if (v > bv) { bv = v; bi = j; }
      }
      for (int off = 16; off > 0; off >>= 1) {
        float ov = __shfl_xor(bv, off, 32);
        int   oi = __shfl_xor(bi, off, 32);
        if (ov > bv || (ov == bv && oi < bi)) { bv = ov; bi = oi; }
      }
      if (!(bv >= THR)) break;                    // cond sticks false forever
      int m = bi;
      if (lane == 0 && m != i) {
        float t;
        t = ss[i];  ss[i]  = ss[m];  ss[m]  = t;
        t = bx1[i]; bx1[i] = bx1[m]; bx1[m] = t;
        t = by1[i]; by1[i] = by1[m]; by1[m] = t;
        t = bx2[i]; bx2[i] = bx2[m]; bx2[m] = t;
        t = by2[i]; by2[i] = by2[m]; by2[m] = t;
      }
      wait_dscnt0();
      float ix1 = bx1[i], iy1 = by1[i], ix2 = bx2[i], iy2 = by2[i];
      float area1 = (ix2 - ix1) * (iy2 - iy1);
      for (int j = i + 1 + lane; j < KK; j += 32) {
        float jx1 = bx1[j], jy1 = by1[j], jx2 = bx2[j], jy2 = by2[j];
        float area2 = (jx2 - jx1) * (jy2 - jy1);
        float lx = fmaxf(ix1, jx1), ly = fmaxf(iy1, jy1);
        float rx = fminf(ix2, jx2), ry = fminf(iy2, jy2);
        float iw = fmaxf(rx - lx, 0.f), ih = fmaxf(ry - ly, 0.f);
        float inter = iw * ih;
        float iou = inter / (area1 + area2 - inter);
        float decay = expf(-iou * iou * SIGMA_INV);
        ss[j] *= decay;
      }
    }
  }
  __syncthreads();

  // ---- outputs: scores | labels(done in A) | boxes | keep ----
  float* out_scores = out;                  // [BK]
  float* out_boxes  = out + 2 * (size_t)BK; // [BK*4]
  float* out_keep   = out + 6 * (size_t)BK; // [BK]
  for (int r = tid; r < KK; r += 512) {
    float sc = ss[r];
    size_t o = (size_t)b * KK + r;
    out_scores[o] = sc;
    size_t ob = o * 4;
    out_boxes[ob + 0] = bx1[r];
    out_boxes[ob + 1] = by1[r];
    out_boxes[ob + 2] = bx2[r];
    out_boxes[ob + 3] = by2[r];
    out_keep[o] = (sc > THR) ? 1.0f : 0.0f;
  }
}

extern "C" void kernel_launch(void* const* d_in, const int* in_sizes, int n_in,
                              void* d_out, int out_size, void* d_ws, size_t ws_size,
                              hipStream_t stream) {
  (void)in_sizes; (void)n_in; (void)out_size; (void)ws_size;
  const float* pred_logits  = (const float*)d_in[0];   // [B,Q,C]
  const float* pred_boxes   = (const float*)d_in[1];   // [B,Q,4]
  const float* target_sizes = (const float*)d_in[2];   // [B,2]
  float* out = (float*)d_out;

  float* ws_scores = (float*)d_ws;           // BK floats
  int*   ws_qidx   = (int*)(ws_scores + BK); // BK ints   (total 614 KB scratch)

  topk_kernel<<<BB, 512, 0, stream>>>(pred_logits, out + (size_t)BK,
                                      ws_scores, ws_qidx);
  nms_kernel<<<BB, 512, 0, stream>>>(pred_boxes, target_sizes,
                                     ws_scores, ws_qidx, out);
}